// FlowLayer_42039139894266
// MI455X (gfx1250) — compile-verified
//
#include <hip/hip_runtime.h>
#include <cstdint>
#include <math.h>

// ---------------- problem constants (match reference) ----------------
#define N_NODES_C 100000
#define N_CH      16
#define DIM       3
#define N_STEPS   2
#define NCD       (N_CH * DIM)            // 48 floats per node
#define FP32_EPS  1.1920929e-7f

constexpr int TILE  = 256;                // edges staged per block
constexpr int BLOCK = 256;                // threads per block (8 waves, wave32)
constexpr int EDGES_PER_PASS = BLOCK / N_CH;   // 16
constexpr int PASSES = TILE / EDGES_PER_PASS;  // 16

typedef uint32_t v4u __attribute__((ext_vector_type(4)));
typedef uint32_t v8u __attribute__((ext_vector_type(8)));

// Guaranteed-native fp32 atomic add, no return value (STOREcnt path, fire-and-
// forget), device scope so the scatter is visible across workgroups.
__device__ __forceinline__ void atomic_add_f32(float* p, float v) {
    asm volatile("global_atomic_add_f32 %0, %1, off scope:SCOPE_DEV"
                 :: "v"((uint64_t)(uintptr_t)p), "v"(v)
                 : "memory");
}

// Tensor Data Mover: 1D tile copy global -> LDS (one instruction, TENSORcnt).
// D# built per CDNA5 ISA 8.3/8.4: group0 = {flags, lds_addr, global_addr, type},
// group1 = {data_size, tensor dims, tile dims, strides}. VADDR2/3 omitted
// (NULL) => <=2D tensor form. tensor_dim0 = elements available from gsrc, so a
// tail tile gets hardware OOB zero-fill instead of reading past the buffer.
__device__ __forceinline__ void tdm_load_1d(uint32_t lds_addr, const void* gsrc,
                                            uint32_t avail_elems) {
    const uint64_t ga = (uint64_t)(uintptr_t)gsrc;
    v4u g0;
    g0[0] = 1u;                                            // count=1, user desc, no gather
    g0[1] = lds_addr;                                      // LDS byte address
    g0[2] = (uint32_t)ga;                                  // global_addr[31:0]
    g0[3] = (uint32_t)((ga >> 32) & 0x01FFFFFFu)           // global_addr[56:32]
          | (2u << 30);                                    // type = 2 ("image")
    v8u g1;
    g1[0] = 2u << 16;                                      // wg_mask=0, data_size=2 (4B)
    g1[1] = (avail_elems & 0xFFFFu) << 16;                 // tensor_dim0[15:0]
    g1[2] = (avail_elems >> 16) | (1u << 16);              // tensor_dim0[31:16], tensor_dim1=1
    g1[3] = ((uint32_t)TILE) << 16;                        // tensor_dim1 hi=0, tile_dim0=TILE
    g1[4] = 0u;                                            // tile_dim1=0, tile_dim2=0 (unused)
    g1[5] = (uint32_t)TILE;                                // tensor_dim0_stride (single row)
    g1[6] = 0u;
    g1[7] = 0u;
    asm volatile("tensor_load_to_lds %0, %1" :: "s"(g0), "s"(g1) : "memory");
}

// ---------------- zero scratch ----------------
__global__ __launch_bounds__(BLOCK) void zero_kernel(float4* __restrict__ p, int n4) {
    int i = blockIdx.x * blockDim.x + threadIdx.x;
    if (i < n4) p[i] = make_float4(0.f, 0.f, 0.f, 0.f);
}

// ---------------- edge kernel: gather + sphere_log + scatter-add ----------------
// Mapping: 16 consecutive lanes = 16 channels of one edge (coalesced 192B node
// blocks). Sender/receiver index tiles staged by the TDM (TENSORcnt); weights
// staged by per-lane async global->LDS copies (ASYNCcnt); both overlap.
template <bool ADD_DEG>
__global__ __launch_bounds__(BLOCK) void edge_kernel(const float* __restrict__ x,
                            const int*   __restrict__ senders,
                            const int*   __restrict__ receivers,
                            const float* __restrict__ weights,
                            float*       __restrict__ agg,
                            float*       __restrict__ deg,
                            int nEdges)
{
    __shared__ int   s_snd[TILE];
    __shared__ int   s_rcv[TILE];
    __shared__ float s_w[TILE];

    const int tid   = threadIdx.x;
    const int tile0 = blockIdx.x * TILE;
    const int idx   = tile0 + tid;

    // Per-lane async staging of weights (ASYNCcnt engine) - issue first so it
    // overlaps the TDM descriptors below.
    if (idx < nEdges) {
        uint32_t l_w = (uint32_t)(uintptr_t)&s_w[tid];
        uint64_t g_w = (uint64_t)(uintptr_t)(weights + idx);
        asm volatile("global_load_async_to_lds_b32 %0, %1, off" :: "v"(l_w), "v"(g_w) : "memory");
    }

    // TDM staging of the index tiles, issued by wave 0 only (wave-uniform
    // guard -> scalar branch; TDM ignores EXEC, and a duplicate issue would be
    // a byte-identical copy anyway).
    if (__builtin_amdgcn_readfirstlane(tid) < 32) {
        const uint32_t remaining = (uint32_t)(nEdges - tile0);
        tdm_load_1d((uint32_t)(uintptr_t)&s_snd[0], senders   + tile0, remaining);
        tdm_load_1d((uint32_t)(uintptr_t)&s_rcv[0], receivers + tile0, remaining);
        asm volatile("s_wait_tensorcnt 0" ::: "memory");
    }
    asm volatile("s_wait_asynccnt 0" ::: "memory");
    __syncthreads();

    const int ch    = tid & (N_CH - 1);
    const int eSub  = tid >> 4;           // 0..15: which edge of the pass
    const int choff = ch * DIM;

    const bool fullTile = (tile0 + TILE) <= nEdges;

    if (fullTile) {
        #pragma unroll 4
        for (int p = 0; p < PASSES; ++p) {
            const int eLoc = p * EDGES_PER_PASS + eSub;

            // Prefetch next pass's random node blocks, one lane per 128B line.
            if (p + 1 < PASSES && (ch & 7) == 0) {
                const int eLocN = eLoc + EDGES_PER_PASS;
                const int sN = s_snd[eLocN];
                const int rN = s_rcv[eLocN];
                const int lineOff = (ch >> 3) * 32;   // 0 or +128 bytes
                __builtin_prefetch(x + sN * NCD + lineOff);
                __builtin_prefetch(x + rN * NCD + lineOff);
            }

            const int   s = s_snd[eLoc];
            const int   r = s_rcv[eLoc];
            const float w = s_w[eLoc];

            const float* xs = x + s * NCD + choff;
            const float* xr = x + r * NCD + choff;
            const float sx = xs[0], sy = xs[1], sz = xs[2];
            const float rx = xr[0], ry = xr[1], rz = xr[2];

            // sphere_log: c = clip(<xs,xr>); theta/sin(theta) * (xr - c*xs)
            float c = sx * rx + sy * ry + sz * rz;
            c = fminf(fmaxf(c, -1.0f + 1e-7f), 1.0f - 1e-7f);
            const float theta = acosf(c);
            // sin(acos(c)) == sqrt(1 - c^2); c is clipped so 1-c^2 >= ~2e-7
            const float fac = theta * rsqrtf(fmaxf(1.0f - c * c, 1e-14f));
            const float fw  = fac * w;

            float* a = agg + s * NCD + choff;
            atomic_add_f32(a + 0, fw * (rx - c * sx));
            atomic_add_f32(a + 1, fw * (ry - c * sy));
            atomic_add_f32(a + 2, fw * (rz - c * sz));
            if (ADD_DEG && ch == 0) atomic_add_f32(deg + s, w);
        }
    } else {
        for (int p = 0; p < PASSES; ++p) {
            const int eLoc = p * EDGES_PER_PASS + eSub;
            const int e    = tile0 + eLoc;
            if (e >= nEdges) continue;

            const int   s = s_snd[eLoc];
            const int   r = s_rcv[eLoc];
            const float w = s_w[eLoc];

            const float* xs = x + s * NCD + choff;
            const float* xr = x + r * NCD + choff;
            const float sx = xs[0], sy = xs[1], sz = xs[2];
            const float rx = xr[0], ry = xr[1], rz = xr[2];

            float c = sx * rx + sy * ry + sz * rz;
            c = fminf(fmaxf(c, -1.0f + 1e-7f), 1.0f - 1e-7f);
            const float theta = acosf(c);
            const float fac = theta * rsqrtf(fmaxf(1.0f - c * c, 1e-14f));
            const float fw  = fac * w;

            float* a = agg + s * NCD + choff;
            atomic_add_f32(a + 0, fw * (rx - c * sx));
            atomic_add_f32(a + 1, fw * (ry - c * sy));
            atomic_add_f32(a + 2, fw * (rz - c * sz));
            if (ADD_DEG && ch == 0) atomic_add_f32(deg + s, w);
        }
    }
}

// ---------------- node kernel: gate + sphere_exp ----------------
// One thread per (node, channel). u = agg/deg = -laplace(x); the reference's
// two sign flips cancel: v_step = u * scale * t.
__global__ __launch_bounds__(BLOCK) void node_kernel(const float* __restrict__ x_in,
                            const float* __restrict__ agg,
                            const float* __restrict__ deg,
                            const float* __restrict__ t_sqrt,
                            const float* __restrict__ a_sqrt,
                            const float* __restrict__ b_sqrt,
                            float*       __restrict__ x_out)
{
    const int tid = blockIdx.x * blockDim.x + threadIdx.x;
    if (tid >= N_NODES_C * N_CH) return;
    const int n = tid >> 4;
    const int c = tid & (N_CH - 1);

    const float ts = t_sqrt[c];
    const float as = a_sqrt[c];
    const float bs = b_sqrt[c];
    const float t     = ts * ts * (1.0f / (float)N_STEPS);
    const float alpha = as * as;
    const float beta  = bs * bs;

    const float dinv = 1.0f / (deg[n] + 1e-12f);
    const int base = tid * DIM;
    const float ux = agg[base + 0] * dinv;
    const float uy = agg[base + 1] * dinv;
    const float uz = agg[base + 2] * dinv;

    const float nrm = sqrtf(ux * ux + uy * uy + uz * uz + FP32_EPS);
    const float z   = beta * (nrm - alpha);
    const float d   = 1.0f / (1.0f + __expf(-z));
    const float scale = (nrm * d <= 1.0f) ? d : (1.0f / nrm);
    const float st = scale * t;
    const float vx = ux * st, vy = uy * st, vz = uz * st;

    // sphere_exp
    const float n2 = sqrtf(vx * vx + vy * vy + vz * vz + 1e-16f);
    const float cn = cosf(n2);
    const float sn = sinf(n2) / n2;

    const float xx = x_in[base + 0], xy = x_in[base + 1], xz = x_in[base + 2];
    x_out[base + 0] = cn * xx + sn * vx;
    x_out[base + 1] = cn * xy + sn * vy;
    x_out[base + 2] = cn * xz + sn * vz;
}

// ---------------- launcher ----------------
extern "C" void kernel_launch(void* const* d_in, const int* in_sizes, int n_in,
                              void* d_out, int out_size, void* d_ws, size_t ws_size,
                              hipStream_t stream)
{
    const float* nodes   = (const float*)d_in[0];
    const int*   senders = (const int*)  d_in[1];
    const int*   recvs   = (const int*)  d_in[2];
    const float* weights = (const float*)d_in[3];
    const float* t_sqrt  = (const float*)d_in[4];
    const float* a_sqrt  = (const float*)d_in[5];
    const float* b_sqrt  = (const float*)d_in[6];
    float* out = (float*)d_out;

    const int nEdges = in_sizes[1];

    // workspace: [agg: N*C*D floats][deg: N floats]  (~19.6 MB)
    float* agg = (float*)d_ws;
    float* deg = agg + (size_t)N_NODES_C * NCD;

    const int aggN   = N_NODES_C * NCD;            // 4,800,000 (mult of 4)
    const int n4_all = (aggN + N_NODES_C) / 4;     // agg + deg
    const int n4_agg = aggN / 4;

    const int nTiles      = (nEdges + TILE - 1) / TILE;
    const int nodeThreads = N_NODES_C * N_CH;
    const int nodeBlocks  = (nodeThreads + BLOCK - 1) / BLOCK;

    // ---- step 1 ----
    zero_kernel<<<(n4_all + BLOCK - 1) / BLOCK, BLOCK, 0, stream>>>((float4*)agg, n4_all);
    edge_kernel<true><<<nTiles, BLOCK, 0, stream>>>(nodes, senders, recvs, weights, agg, deg, nEdges);
    node_kernel<<<nodeBlocks, BLOCK, 0, stream>>>(nodes, agg, deg, t_sqrt, a_sqrt, b_sqrt, out);

    // ---- step 2 (deg is invariant; reuse it) ----
    zero_kernel<<<(n4_agg + BLOCK - 1) / BLOCK, BLOCK, 0, stream>>>((float4*)agg, n4_agg);
    edge_kernel<false><<<nTiles, BLOCK, 0, stream>>>(out, senders, recvs, weights, agg, deg, nEdges);
    node_kernel<<<nodeBlocks, BLOCK, 0, stream>>>(out, agg, deg, t_sqrt, a_sqrt, b_sqrt, out);
}